// BSplineActivation_48653389529324
// MI455X (gfx1250) — compile-verified
//
#include <hip/hip_runtime.h>
#include <hip/hip_bf16.h>

// B-spline activation, NCHW f32. Memory-bound (AI ~1 FLOP/B): target is the
// 23.3 TB/s HBM roofline (~17.6 us for 411 MB). One block per (n,c) slab of
// 3136 contiguous elements; the channel's 51-float coefficient row is staged
// into LDS via the CDNA5 async global->LDS path, gathers become ds_read2_b32,
// and the main stream uses non-temporal b128 loads/stores (dataset > 192 MB L2).

#define SIZE_K 51    // knots per activation
#define HALF_K 25    // SIZE_K / 2
#define CCH    256   // channels
#define HW     3136  // 56*56, contiguous per (n,c)
#define VECS   784   // HW / 4

typedef __attribute__((ext_vector_type(4))) float float4v;

typedef __attribute__((address_space(1))) int as1_int;
typedef __attribute__((address_space(3))) int as3_int;

#if defined(__HIP_DEVICE_COMPILE__) && __has_builtin(__builtin_amdgcn_global_load_async_to_lds_b32)
#define USE_ASYNC_LDS 1
#else
#define USE_ASYNC_LDS 0
#endif

__device__ __forceinline__ void wait_async_zero() {
#if __has_builtin(__builtin_amdgcn_s_wait_asynccnt)
    __builtin_amdgcn_s_wait_asynccnt(0);
#else
    asm volatile("s_wait_asynccnt 0" ::: "memory");
#endif
}

__global__ __launch_bounds__(256) void bspline_act_kernel(
    const float* __restrict__ x,
    const float* __restrict__ coef,
    float* __restrict__ out)
{
    __shared__ float lds_coef[SIZE_K + 1];   // 52 floats; index up to 50 used

    const int tid  = threadIdx.x;
    const int blk  = blockIdx.x;             // blk = n*C + c
    const int c    = blk & (CCH - 1);
    const long long base  = (long long)blk * HW;
    const int       cbase = c * SIZE_K;

    // ---- Stage this channel's coefficient row into LDS (CDNA5 async copy) ----
    if (tid < SIZE_K) {
#if USE_ASYNC_LDS
        __builtin_amdgcn_global_load_async_to_lds_b32(
            (as1_int*)(coef + cbase + tid),
            (as3_int*)(&lds_coef[tid]),
            /*imm offset=*/0, /*cpol=*/0);
#else
        lds_coef[tid] = coef[cbase + tid];
#endif
    }
#if USE_ASYNC_LDS
    wait_async_zero();                       // ASYNCcnt -> 0 before barrier
#endif
    __syncthreads();

    // ---- Stream 3136 floats as 784 float4s (b128, non-temporal) ----
    const float4v* __restrict__ xin  = (const float4v*)(x + base);
    float4v*       __restrict__ xout = (float4v*)(out + base);

    for (int i = tid; i < VECS; i += 256) {
        float4v v = __builtin_nontemporal_load(&xin[i]);
        float4v r;
#pragma unroll
        for (int k = 0; k < 4; ++k) {
            float xv = v[k];
            // clamp so idx and idx+1 stay inside this channel's knot range
            float xc = fminf(fmaxf(xv, -2.5f), 2.4f);
            float fl = __builtin_floorf(xc * 10.0f);          // left-knot offset
            float fr = __builtin_fmaf(xv, 10.0f, -fl);        // frac uses UNclamped x
            int   li = (int)fl + HALF_K;                      // 0..49
            float c0 = lds_coef[li];
            float c1 = lds_coef[li + 1];                      // -> ds_read2_b32 pair
            r[k] = __builtin_fmaf(fr, c1 - c0, c0);           // lerp
        }
        __builtin_nontemporal_store(r, &xout[i]);
    }
}

extern "C" void kernel_launch(void* const* d_in, const int* in_sizes, int n_in,
                              void* d_out, int out_size, void* d_ws, size_t ws_size,
                              hipStream_t stream) {
    (void)in_sizes; (void)n_in; (void)d_ws; (void)ws_size; (void)out_size;
    const float* x    = (const float*)d_in[0];   // [N,C,H,W] f32
    const float* coef = (const float*)d_in[1];   // [C*SIZE] f32
    // d_in[2] (zero_knot_indexes) is implied by c*SIZE + HALF_K; unused.
    float* out = (float*)d_out;

    const int blocks = 64 * CCH;                 // N*C = 16384 (one per (n,c))
    bspline_act_kernel<<<blocks, 256, 0, stream>>>(x, coef, out);
}